// toy_che_24945170055809
// MI455X (gfx1250) — compile-verified
//
#include <hip/hip_runtime.h>
#include <stdint.h>

// Problem geometry (fixed by the reference): 16 planes of 2048x2048 f32,
// periodic boundary conditions, D = 0.01.
#define WDIM   2048
#define WMASK  2047
#define TX     128          // output tile width
#define TY     32           // output tile height
#define HX     4            // x halo (logical need 2; padded to 4 for b128 alignment)
#define HY     2            // y halo
#define AW     (TX + 2*HX)  // 136 floats per staged row
#define AH     (TY + 2*HY)  // 36 staged rows
#define ASTR   AW           // 136; row pitch = 544 B = 34*16 -> every float4 slot 16B aligned
#define IW     (TX + 2)     // 130
#define IH     (TY + 2)     // 34
#define ISTR   132          // inner-tile pitch (padded)
#define NLOAD  ((AW/4)*AH)  // 1224 float4 async-load tasks
#define D_COEF 0.01f

typedef __attribute__((ext_vector_type(4))) float v4f;  // native vector for NT store

// Low 32 bits of a flat pointer into LDS == LDS byte offset (aperture layout:
// addr[63:32] = shared aperture base, addr[31:0] = wave-relative LDS offset).
__device__ __forceinline__ uint32_t lds_off(const void* p) {
    return (uint32_t)(uintptr_t)p;
}

__global__ __launch_bounds__(256)
void che_stencil_fused(const float* __restrict__ a, float* __restrict__ out) {
    __shared__ float sA[AH * ASTR];   // staged a-tile (with halo)
    __shared__ float sI[IH * ISTR];   // inner = a^3 - a - Lap(a)

    const int tid = threadIdx.x;
    const int tx0 = blockIdx.x * TX;
    const int ty0 = blockIdx.y * TY;
    const size_t plane = (size_t)blockIdx.z * (size_t)(WDIM * WDIM);
    const float* __restrict__ ap = a + plane;
    float* __restrict__ op = out + plane;

    // ---- Phase 1: async global -> LDS staging of the haloed a-tile -------
    // 34 float4 chunks per row x 36 rows = 1224 tasks over 256 lanes.
    // These are GLOBAL_LOAD_ASYNC_TO_LDS_B128 (ASYNCcnt-tracked, no VGPR
    // round-trip); tail lanes duplicate the last chunk so EXEC stays full.
    #pragma unroll
    for (int k = 0; k < 5; ++k) {
        int idx = tid + k * 256;
        if (idx >= NLOAD) idx = NLOAD - 1;
        int r = idx / (AW / 4);
        int c = idx - r * (AW / 4);
        int gy = (ty0 - HY + r + WDIM) & WMASK;     // periodic wrap (2048 = 2^11)
        int gx = (tx0 - HX + 4 * c + WDIM) & WMASK; // stays float4-aligned after wrap
        const float* gsrc = ap + (size_t)gy * WDIM + gx;
        uint32_t dst = lds_off(&sA[r * ASTR + 4 * c]);
        asm volatile("global_load_async_to_lds_b128 %0, %1, off"
                     :: "v"(dst), "v"(gsrc) : "memory");
    }
    asm volatile("s_wait_asynccnt 0" ::: "memory");
    __syncthreads();

    // ---- Phase 2: inner = a^3 - a - Lap(a) over (130 x 34) ---------------
    for (int k = 0; k < 18; ++k) {
        int idx = tid + k * 256;
        if (idx < IW * IH) {
            int iy = idx / IW;
            int ix = idx - iy * IW;
            const float* p = &sA[(iy + HY - 1) * ASTR + (ix + HX - 1)];
            float c   = p[0];
            float lap = p[-1] + p[1] + p[-ASTR] + p[ASTR] - 4.0f * c;
            sI[iy * ISTR + ix] = c * c * c - c - lap;
        }
    }
    __syncthreads();

    // ---- Phase 3: out = a + D * Lap(inner), non-temporal float4 stores ---
    // Output is write-once / never re-read on device: NT keeps the 192MB L2
    // for the input stream (halo rows get cross-workgroup reuse there).
    #pragma unroll
    for (int k = 0; k < 4; ++k) {
        int idx = tid + k * 256;      // 0..1023 : 32 rows x 32 float4-chunks
        int oy  = idx >> 5;
        int ox  = (idx & 31) * 4;
        v4f v;
        #pragma unroll
        for (int j = 0; j < 4; ++j) {
            int x = ox + j;
            const float* ip = &sI[(oy + 1) * ISTR + (x + 1)];
            float lap = ip[-1] + ip[1] + ip[-ISTR] + ip[ISTR] - 4.0f * ip[0];
            v[j] = sA[(oy + HY) * ASTR + (x + HX)] + D_COEF * lap;
        }
        __builtin_nontemporal_store(v, (v4f*)(op + (size_t)(ty0 + oy) * WDIM + (tx0 + ox)));
    }
}

extern "C" void kernel_launch(void* const* d_in, const int* in_sizes, int n_in,
                              void* d_out, int out_size, void* d_ws, size_t ws_size,
                              hipStream_t stream) {
    const float* a = (const float*)d_in[0];
    float* out = (float*)d_out;
    int nplanes = in_sizes[0] / (WDIM * WDIM);   // 16 for the reference shapes
    dim3 grid(WDIM / TX, WDIM / TY, nplanes);
    che_stencil_fused<<<grid, 256, 0, stream>>>(a, out);
}